// GCN_35974646071652
// MI455X (gfx1250) — compile-verified
//
#include <hip/hip_runtime.h>
#include <hip/hip_bf16.h>

// GCN: B=4 graphs, N=50000 nodes, E=800000 edges, F=64 features, 2 GCNConv layers.
// f32 throughout (matches reference). GEMM via V_WMMA_F32_16X16X4_F32.

#define BG 4
#define NN 50000
#define EE 800000
#define FF 64
#define NTILES (NN / 16)   // 3125 row tiles per graph

typedef float v2f __attribute__((ext_vector_type(2)));
typedef float v8f __attribute__((ext_vector_type(8)));

// ---------------------------------------------------------------- utilities
__global__ void k_zero(float* __restrict__ p, int n) {
    int i = blockIdx.x * blockDim.x + threadIdx.x;
    if (i < n) p[i] = 0.0f;
}

// deg[b*N + dst] += ew  (one thread per edge)
__global__ void k_deg(const int* __restrict__ ei, const float* __restrict__ ea,
                      float* __restrict__ deg) {
    unsigned i = blockIdx.x * blockDim.x + threadIdx.x;
    if (i >= (unsigned)BG * EE) return;
    unsigned b = i / EE, e = i % EE;
    int dst = ei[(size_t)b * 2 * EE + EE + e];
    atomicAdd(&deg[(size_t)b * NN + dst], ea[(size_t)b * EE + e]);
}

// dinv = rsqrt(deg + 1)   (self-loop adds 1, so deg >= 1 always)
__global__ void k_dinv(float* __restrict__ d, int n) {
    int i = blockIdx.x * blockDim.x + threadIdx.x;
    if (i < n) d[i] = rsqrtf(d[i] + 1.0f);
}

// ---------------------------------------------------------------- WMMA GEMM
// H[b] = X[b] @ W.  X: [N,64] row-major, W: [64,64] row-major, H: [N,64].
// One wave32 per 16-row tile; 4 column tiles of 16; K = 64 as 16 steps of 4.
__global__ __launch_bounds__(128) void k_gemm(const float* __restrict__ X,
                                              const float* __restrict__ W,
                                              float* __restrict__ H) {
    int lane = threadIdx.x & 31;
    int wave = threadIdx.x >> 5;
    int gw   = blockIdx.x * 4 + wave;          // global wave id
    if (gw >= BG * NTILES) return;             // uniform per-wave branch
    int b    = gw / NTILES;
    int tile = gw % NTILES;

    const float* x = X + (size_t)b * NN * FF;
    float*       h = H + (size_t)b * NN * FF;

    int r0   = tile * 16;
    int half = lane >> 4;      // 0: lanes 0-15, 1: lanes 16-31
    int m    = lane & 15;

    v8f acc[4] = {v8f{}, v8f{}, v8f{}, v8f{}};

    for (int kb = 0; kb < 16; ++kb) {          // K blocks of 4
        // A fragment: row (r0+m); v0 = K(half*2), v1 = K(half*2+1)
        const float* ap = x + (size_t)(r0 + m) * FF + kb * 4 + half * 2;
        v2f afrag;
        afrag.x = ap[0];
        afrag.y = ap[1];
#pragma unroll
        for (int c = 0; c < 4; ++c) {          // 4 column tiles of 16
            // B fragment: col (c*16+m); v0 = K(half*2) row of W, v1 = next row
            const float* bp = W + (size_t)(kb * 4 + half * 2) * FF + c * 16 + m;
            v2f bfrag;
            bfrag.x = bp[0];
            bfrag.y = bp[FF];
            acc[c] = __builtin_amdgcn_wmma_f32_16x16x4_f32(
                false, afrag, false, bfrag, (short)0, acc[c], false, false);
        }
    }
#pragma unroll
    for (int c = 0; c < 4; ++c)
#pragma unroll
        for (int j = 0; j < 8; ++j)
            h[(size_t)(r0 + j + 8 * half) * FF + c * 16 + m] = acc[c][j];
}

// ---------------------------------------------------------------- self loop
// out = h * dinv^2   (float4 vectorized; initializes the accumulator)
__global__ void k_self(const float* __restrict__ H, const float* __restrict__ dinv,
                       float* __restrict__ out) {
    unsigned i = blockIdx.x * blockDim.x + threadIdx.x;
    const unsigned total = (unsigned)BG * NN * (FF / 4);
    if (i >= total) return;
    unsigned node = i / (FF / 4);              // global node index b*N+n
    float d = dinv[node];
    float s = d * d;
    float4 v = ((const float4*)H)[i];
    v.x *= s; v.y *= s; v.z *= s; v.w *= s;
    ((float4*)out)[i] = v;
}

// ---------------------------------------------------------------- edge scatter
// 16 threads per edge; thread j handles features [4j, 4j+3].
// out[dst] += h[src] * (dinv[src] * ew * dinv[dst])
__global__ void k_edge(const int* __restrict__ ei, const float* __restrict__ ea,
                       const float* __restrict__ dinv, const float* __restrict__ H,
                       float* __restrict__ out) {
    unsigned i = blockIdx.x * blockDim.x + threadIdx.x;
    const unsigned total = (unsigned)BG * EE * 16;
    if (i >= total) return;
    unsigned j  = i & 15;
    unsigned eg = i >> 4;                      // 0 .. B*E-1
    unsigned b  = eg / EE, e = eg % EE;

    int   src = ei[(size_t)b * 2 * EE + e];
    int   dst = ei[(size_t)b * 2 * EE + EE + e];
    float w   = ea[(size_t)b * EE + e];
    float nrm = dinv[(size_t)b * NN + src] * w * dinv[(size_t)b * NN + dst];

    float4 v = ((const float4*)(H + ((size_t)b * NN + src) * FF))[j];
    float* op = out + ((size_t)b * NN + dst) * FF + j * 4;
    atomicAdd(op + 0, v.x * nrm);
    atomicAdd(op + 1, v.y * nrm);
    atomicAdd(op + 2, v.z * nrm);
    atomicAdd(op + 3, v.w * nrm);
}

// ---------------------------------------------------------------- bias + ReLU
__global__ void k_bias_relu(float* __restrict__ out, const float* __restrict__ bias) {
    unsigned i = blockIdx.x * blockDim.x + threadIdx.x;
    const unsigned total = (unsigned)BG * NN * (FF / 4);
    if (i >= total) return;
    unsigned f4 = i % (FF / 4);
    float4 bv = ((const float4*)bias)[f4];
    float4 v  = ((float4*)out)[i];
    v.x = fmaxf(v.x + bv.x, 0.0f);
    v.y = fmaxf(v.y + bv.y, 0.0f);
    v.z = fmaxf(v.z + bv.z, 0.0f);
    v.w = fmaxf(v.w + bv.w, 0.0f);
    ((float4*)out)[i] = v;
}

// ---------------------------------------------------------------- launcher
extern "C" void kernel_launch(void* const* d_in, const int* in_sizes, int n_in,
                              void* d_out, int out_size, void* d_ws, size_t ws_size,
                              hipStream_t stream) {
    (void)in_sizes; (void)n_in; (void)out_size; (void)ws_size;

    const float* x  = (const float*)d_in[0];   // [B,N,F]
    const int*   ei = (const int*)d_in[1];     // [B,2,E]
    const float* ea = (const float*)d_in[2];   // [B,E]
    const float* W1 = (const float*)d_in[3];   // [F,F]
    const float* b1 = (const float*)d_in[4];   // [F]
    const float* W2 = (const float*)d_in[5];   // [F,F]
    const float* b2 = (const float*)d_in[6];   // [F]
    float* out = (float*)d_out;                // [B,N,F]

    // workspace layout (floats): dinv [B*N] | h [B*N*F]
    float* ws   = (float*)d_ws;
    size_t off  = ((size_t)BG * NN + 63) & ~(size_t)63;
    float* dinv = ws;
    float* h    = ws + off;

    const int T = 256;
    const int nBN   = BG * NN;                        // 200000
    const int nBE   = BG * EE;                        // 3.2M
    const int nBNF4 = BG * NN * (FF / 4);             // 3.2M
    const unsigned nEdgeT = (unsigned)BG * EE * 16;   // 51.2M

    // degree -> dinv (shared by both layers)
    k_zero<<<(nBN + T - 1) / T, T, 0, stream>>>(dinv, nBN);
    k_deg<<<(nBE + T - 1) / T, T, 0, stream>>>(ei, ea, dinv);
    k_dinv<<<(nBN + T - 1) / T, T, 0, stream>>>(dinv, nBN);

    const int gemmBlocks = (BG * NTILES + 3) / 4;     // 4 waves per block

    // ---- layer 1: out = relu( scatter(x@W1) + (x@W1)*dinv^2 + b1 )
    k_gemm<<<gemmBlocks, 128, 0, stream>>>(x, W1, h);
    k_self<<<(nBNF4 + T - 1) / T, T, 0, stream>>>(h, dinv, out);
    k_edge<<<(nEdgeT + T - 1) / T, T, 0, stream>>>(ei, ea, dinv, h, out);
    k_bias_relu<<<(nBNF4 + T - 1) / T, T, 0, stream>>>(out, b1);

    // ---- layer 2: same, input = layer-1 output (read fully into h first)
    k_gemm<<<gemmBlocks, 128, 0, stream>>>(out, W2, h);
    k_self<<<(nBNF4 + T - 1) / T, T, 0, stream>>>(h, dinv, out);
    k_edge<<<(nEdgeT + T - 1) / T, T, 0, stream>>>(ei, ea, dinv, h, out);
    k_bias_relu<<<(nBNF4 + T - 1) / T, T, 0, stream>>>(out, b2);
}